// MemoryInterface_25718264168586
// MI455X (gfx1250) — compile-verified
//
#include <hip/hip_runtime.h>
#include <math.h>

typedef float v2f __attribute__((ext_vector_type(2)));
typedef float v8f __attribute__((ext_vector_type(8)));
typedef unsigned int v4u __attribute__((ext_vector_type(4)));
typedef int v8i __attribute__((ext_vector_type(8)));
typedef int v4i __attribute__((ext_vector_type(4)));

#define DDIM 128
#define CDIM 512
#define BDIM 64
#define TOPK 8
#define CHUNK 128
#define DP 132          // padded LDS row stride in floats (TDM: pad 4 dwords every 128)
#define NBLK2 100       // grid for the similarity kernel

#ifndef __has_builtin
#define __has_builtin(x) 0
#endif

// Only the DEVICE pass declares AMDGPU builtins; enforce TDM availability there.
// The host pass parses the fallback loader (never executed on device).
#if defined(__HIP_DEVICE_COMPILE__)
#if !__has_builtin(__builtin_amdgcn_tensor_load_to_lds)
#error "gfx1250 TDM builtin __builtin_amdgcn_tensor_load_to_lds not available (device pass)"
#endif
#if !__has_builtin(__builtin_amdgcn_s_wait_tensorcnt)
#error "gfx1250 builtin __builtin_amdgcn_s_wait_tensorcnt not available (device pass)"
#endif
#define USE_TDM 1
#else
#define USE_TDM 0
#endif

// ------------- kernel 1: qn = l2norm(query @ Wq + bq) -------------
__global__ void k_qproj(const float* __restrict__ query,
                        const float* __restrict__ Wq,
                        const float* __restrict__ bq,
                        float* __restrict__ qn) {
  __shared__ float qrow[CDIM];
  __shared__ float red[DDIM];
  const int b = blockIdx.x, d = threadIdx.x;
  for (int c = d; c < CDIM; c += DDIM) qrow[c] = query[b * CDIM + c];
  __syncthreads();
  float acc = bq[d];
#pragma unroll 8
  for (int c = 0; c < CDIM; ++c) acc = fmaf(qrow[c], Wq[c * DDIM + d], acc);
  red[d] = acc * acc;
  __syncthreads();
  for (int s = DDIM / 2; s > 0; s >>= 1) {
    if (d < s) red[d] += red[d + s];
    __syncthreads();
  }
  const float rn = 1.0f / fmaxf(sqrtf(red[0]), 1e-12f);
  qn[b * DDIM + d] = acc * rn;
}

#if USE_TDM
// Issue one TDM descriptor: load a CHUNK x 128 f32 tile of `keys` starting at
// row `row0` into LDS at byte offset lds_off, with 4-dword padding appended to
// every 128-dword row (row stride DP=132 floats). OOB rows (>= N) read zero.
__device__ __forceinline__ void tdm_load_tile(unsigned lds_off,
                                              const float* __restrict__ keys,
                                              int row0, int N) {
  const unsigned long long ga =
      (unsigned long long)(uintptr_t)(keys + (size_t)row0 * DDIM);
  // group0: count=1 | lds_addr | global_addr[120:64] | type=2
  const v4u g0 = {1u, lds_off, (unsigned)(ga & 0xFFFFFFFFu),
                  (unsigned)(((ga >> 32) & 0x01FFFFFFu) | (2u << 30))};
  // group1:
  // w0: data_size=2 (4B) | pad_enable | pad_interval=6 (128 dw) | pad_amount=3 (4 dw)
  // w1: tensor_dim0[15:0]=128 at bits 31:16
  // w2: tensor_dim1[15:0]=N.lo16 at bits 31:16
  // w3: tensor_dim1[31:16]=N.hi16 | tile_dim0=128 at bits 31:16
  // w4: tile_dim1=128 (tile_dim2=0)
  // w5: tensor_dim0_stride=128 ; w6,w7: 0
  const v8i g1 = {(int)0x07920000,
                  (int)(128u << 16),
                  (int)(((unsigned)N & 0xFFFFu) << 16),
                  (int)((((unsigned)N >> 16) & 0xFFFFu) | (128u << 16)),
                  (int)128,
                  (int)128,
                  0,
                  0};
  const v4i z4 = {0, 0, 0, 0};
#if defined(__clang_major__) && __clang_major__ >= 23
  const v8i z8 = {0, 0, 0, 0, 0, 0, 0, 0};
  __builtin_amdgcn_tensor_load_to_lds(g0, g1, z4, z4, z8, 0);
#else
  __builtin_amdgcn_tensor_load_to_lds(g0, g1, z4, z4, 0);
#endif
}
#endif

// ------------- kernel 2: TDM-fed WMMA similarity + per-block top-k -------------
// sim[b][n] = dot(qn[b], keys[n]) * importance[n] / max(||keys[n]||, eps)
__global__ void k_sim_topk(const float* __restrict__ keys,
                           const float* __restrict__ importance,
                           const float* __restrict__ qn,
                           float* __restrict__ cand_val,
                           int* __restrict__ cand_idx,
                           int N) {
  extern __shared__ float smem[];
  float* qbuf = smem;                        // BDIM * DP
  float* kb0  = qbuf + BDIM * DP;            // CHUNK * DP (tile buffer 0)
  float* kb1  = kb0 + CHUNK * DP;            // CHUNK * DP (tile buffer 1)
  float* sbuf = kb1 + CHUNK * DP;            // BDIM * DP
  float* rs   = sbuf + BDIM * DP;            // CHUNK

  const int tid  = threadIdx.x;              // 0..255, 8 waves of 32
  const int lane = tid & 31;
  const int w    = tid >> 5;                 // wave id: owns N-subtile w
  const int half = lane >> 4;                // K-pair select per WMMA layout
  const int lr   = lane & 15;

  // stage qn (64x128) into LDS once
  for (int i = tid; i < BDIM * (DDIM / 4); i += 256) {
    const int r = i >> 5, c4 = (i & 31) << 2;
    const float4 v = *(const float4*)(qn + r * DDIM + c4);
    *(float4*)(qbuf + r * DP + c4) = v;
  }

  float tv[TOPK];
  int ti[TOPK];
#pragma unroll
  for (int k = 0; k < TOPK; ++k) { tv[k] = -INFINITY; ti[k] = -1; }
  float vmin = -INFINITY;
  int minpos = 0;

  const int nChunks = (N + CHUNK - 1) / CHUNK;

#if USE_TDM
  // prologue: DMA first tile into buffer 0 (wave 0 issues; TENSORcnt tracks it)
  if (tid < 32 && (int)blockIdx.x < nChunks)
    tdm_load_tile((unsigned)(uintptr_t)kb0, keys, blockIdx.x * CHUNK, N);
#endif

  int cur = 0;
  for (int chunk = blockIdx.x; chunk < nChunks; chunk += gridDim.x) {
    const int base = chunk * CHUNK;
    float* kbuf;
#if USE_TDM
    kbuf = cur ? kb1 : kb0;
    {
      const int nxt = chunk + (int)gridDim.x;
      if (tid < 32) {
        if (nxt < nChunks) {  // pipeline: issue next tile, wait for current
          tdm_load_tile((unsigned)(uintptr_t)(cur ? kb0 : kb1), keys,
                        nxt * CHUNK, N);
          __builtin_amdgcn_s_wait_tensorcnt(1);
        } else {
          __builtin_amdgcn_s_wait_tensorcnt(0);
        }
      }
    }
    __syncthreads();  // current tile visible to all waves
#else
    // host-pass / non-TDM fallback: synchronous staged copy
    kbuf = kb0;
    __syncthreads();
    for (int i = tid; i < CHUNK * (DDIM / 4); i += 256) {
      const int r = i >> 5, c4 = (i & 31) << 2;
      const int n = base + r;
      float4 v = make_float4(0.f, 0.f, 0.f, 0.f);
      if (n < N) v = *(const float4*)(keys + (size_t)n * DDIM + c4);
      *(float4*)(kbuf + r * DP + c4) = v;
    }
    __syncthreads();
#endif

    // per-key scale = importance / max(||key||, eps)
    if (tid < CHUNK) {
      const int n = base + tid;
      float s2 = 0.f;
#pragma unroll 16
      for (int j = 0; j < DDIM; ++j) {
        const float x = kbuf[tid * DP + j];
        s2 = fmaf(x, x, s2);
      }
      const float imp = (n < N) ? importance[n] : 0.f;
      rs[tid] = imp / fmaxf(sqrtf(s2), 1e-12f);
    }
    __syncthreads();

    // WMMA: 64x128 sim tile = 4 M-subtiles x 8 N-subtiles (one N-subtile/wave)
    v8f acc0 = {}, acc1 = {}, acc2 = {}, acc3 = {};
    const int nl = w * 16 + lr;  // this lane's key column within the chunk
#pragma unroll 4
    for (int ks = 0; ks < DDIM / 4; ++ks) {
      const int k = 4 * ks + 2 * half;
      // B (4x16): element (K,N) = keys[n][K]; lanes 0-15: K=k..k+1, 16-31: K=k+2..k+3
      const v2f bf = *(const v2f*)(kbuf + nl * DP + k);
      const v2f a0 = *(const v2f*)(qbuf + (0 + lr) * DP + k);
      const v2f a1 = *(const v2f*)(qbuf + (16 + lr) * DP + k);
      const v2f a2 = *(const v2f*)(qbuf + (32 + lr) * DP + k);
      const v2f a3 = *(const v2f*)(qbuf + (48 + lr) * DP + k);
      acc0 = __builtin_amdgcn_wmma_f32_16x16x4_f32(false, a0, false, bf, (short)0, acc0, false, false);
      acc1 = __builtin_amdgcn_wmma_f32_16x16x4_f32(false, a1, false, bf, (short)0, acc1, false, false);
      acc2 = __builtin_amdgcn_wmma_f32_16x16x4_f32(false, a2, false, bf, (short)0, acc2, false, false);
      acc3 = __builtin_amdgcn_wmma_f32_16x16x4_f32(false, a3, false, bf, (short)0, acc3, false, false);
    }
    // C layout: vgpr v, lane l -> M = v + 8*(l>>4), N = l&15
    const float sc = rs[nl];
#pragma unroll
    for (int v = 0; v < 8; ++v) {
      const int mrow = v + 8 * half;
      sbuf[(mrow)*DP + nl]        = acc0[v] * sc;
      sbuf[(mrow + 16) * DP + nl] = acc1[v] * sc;
      sbuf[(mrow + 32) * DP + nl] = acc2[v] * sc;
      sbuf[(mrow + 48) * DP + nl] = acc3[v] * sc;
    }
    __syncthreads();

    // top-k scan: one thread per batch row
    if (tid < BDIM) {
      const int limit = (N - base < CHUNK) ? (N - base) : CHUNK;
      for (int j = 0; j < limit; ++j) {
        const float v = sbuf[tid * DP + j];
        if (v > vmin) {
#pragma unroll
          for (int k = 0; k < TOPK; ++k)
            if (k == minpos) { tv[k] = v; ti[k] = base + j; }
          vmin = tv[0]; minpos = 0;
#pragma unroll
          for (int k = 1; k < TOPK; ++k)
            if (tv[k] < vmin) { vmin = tv[k]; minpos = k; }
        }
      }
    }
    __syncthreads();  // all reads of kbuf/sbuf done before next overwrite
    cur ^= 1;
  }

  if (tid < BDIM) {
    const int o = (blockIdx.x * BDIM + tid) * TOPK;
#pragma unroll
    for (int k = 0; k < TOPK; ++k) { cand_val[o + k] = tv[k]; cand_idx[o + k] = ti[k]; }
  }
}

// ------------- kernel 3: merge per-block candidates into final top-8 -------------
__global__ void k_merge(const float* __restrict__ cand_val,
                        const int* __restrict__ cand_idx,
                        int* __restrict__ topidx, int nblocks) {
  const int t = threadIdx.x;  // one thread per batch row (64)
  float tv[TOPK];
  int ti[TOPK];
#pragma unroll
  for (int k = 0; k < TOPK; ++k) { tv[k] = -INFINITY; ti[k] = -1; }
  float vmin = -INFINITY;
  int minpos = 0;
  for (int bb = 0; bb < nblocks; ++bb) {
    const int o = (bb * BDIM + t) * TOPK;
#pragma unroll
    for (int k = 0; k < TOPK; ++k) {
      const float v = cand_val[o + k];
      const int id = cand_idx[o + k];
      if (id >= 0 && v > vmin) {
#pragma unroll
        for (int q = 0; q < TOPK; ++q)
          if (q == minpos) { tv[q] = v; ti[q] = id; }
        vmin = tv[0]; minpos = 0;
#pragma unroll
        for (int q = 1; q < TOPK; ++q)
          if (tv[q] < vmin) { vmin = tv[q]; minpos = q; }
      }
    }
  }
#pragma unroll
  for (int k = 0; k < TOPK; ++k) topidx[t * TOPK + k] = (ti[k] < 0) ? 0 : ti[k];
}

// ------------- kernel 4: gather + softmax attention + output projection -------------
__global__ void k_attn_out(const float* __restrict__ values,
                           const int* __restrict__ topidx,
                           const float* __restrict__ Wa,
                           const float* __restrict__ ba,
                           const float* __restrict__ Wc,
                           const float* __restrict__ bc,
                           float* __restrict__ out) {
  __shared__ float red[DDIM];
  __shared__ float logits[TOPK];
  __shared__ float comb[DDIM];
  const int b = blockIdx.x, d = threadIdx.x;
  float rv[TOPK];
#pragma unroll
  for (int k = 0; k < TOPK; ++k) {
    const int idx = topidx[b * TOPK + k];
    rv[k] = values[(size_t)idx * DDIM + d];
  }
  const float wa = Wa[d];
  for (int k = 0; k < TOPK; ++k) {
    red[d] = rv[k] * wa;
    __syncthreads();
    for (int s = DDIM / 2; s > 0; s >>= 1) {
      if (d < s) red[d] += red[d + s];
      __syncthreads();
    }
    if (d == 0) logits[k] = red[0] + ba[0];
    __syncthreads();
  }
  float m = logits[0];
#pragma unroll
  for (int k = 1; k < TOPK; ++k) m = fmaxf(m, logits[k]);
  float e[TOPK], s = 0.f;
#pragma unroll
  for (int k = 0; k < TOPK; ++k) { e[k] = expf(logits[k] - m); s += e[k]; }
  const float inv = 1.f / s;
  float c = 0.f;
#pragma unroll
  for (int k = 0; k < TOPK; ++k) c = fmaf(e[k] * inv, rv[k], c);
  comb[d] = c;
  __syncthreads();
  float o = bc[d];
#pragma unroll 8
  for (int dd = 0; dd < DDIM; ++dd) o = fmaf(comb[dd], Wc[dd * DDIM + d], o);
  out[b * DDIM + d] = o;
}

extern "C" void kernel_launch(void* const* d_in, const int* in_sizes, int n_in,
                              void* d_out, int out_size, void* d_ws, size_t ws_size,
                              hipStream_t stream) {
  const float* query      = (const float*)d_in[0];
  const float* keys       = (const float*)d_in[1];
  const float* values     = (const float*)d_in[2];
  const float* importance = (const float*)d_in[3];
  const float* Wq         = (const float*)d_in[4];
  const float* bq         = (const float*)d_in[5];
  const float* Wa         = (const float*)d_in[6];
  const float* ba         = (const float*)d_in[7];
  const float* Wc         = (const float*)d_in[8];
  const float* bc         = (const float*)d_in[9];
  const int N = in_sizes[3];  // 200000 keys
  float* out = (float*)d_out;

  // workspace layout
  float* qn       = (float*)d_ws;                   // 64*128 floats
  float* cand_val = qn + BDIM * DDIM;               // NBLK2*64*8 floats
  int*   cand_idx = (int*)(cand_val + NBLK2 * BDIM * TOPK);
  int*   topidx   = cand_idx + NBLK2 * BDIM * TOPK; // 64*8 ints

  k_qproj<<<BDIM, DDIM, 0, stream>>>(query, Wq, bq, qn);

  const size_t smem2 =
      (size_t)(BDIM * DP + 2 * CHUNK * DP + BDIM * DP + CHUNK) * sizeof(float);
  k_sim_topk<<<NBLK2, 256, smem2, stream>>>(keys, importance, qn, cand_val,
                                            cand_idx, N);

  k_merge<<<1, BDIM, 0, stream>>>(cand_val, cand_idx, topidx, NBLK2);

  k_attn_out<<<BDIM, DDIM, 0, stream>>>(values, topidx, Wa, ba, Wc, bc, out);
}